// NeRFRenderer_72507637891205
// MI455X (gfx1250) — compile-verified
//
#include <hip/hip_runtime.h>
#include <hip/hip_bf16.h>
#include <math.h>

// ---------------------------------------------------------------------------
// NeRF-style volume renderer for MI455X (gfx1250).
// Bandwidth-bound (~268MB streamed + ~34MB L2-resident grid gathers):
// strategy = wave32-per-ray, async global->LDS double buffering (ASYNCcnt),
// wave shuffle scan for the exclusive transmittance cumsum.
// ---------------------------------------------------------------------------

#define NSTEPS   128
#define CHUNK    32            // steps per wave pass (== wave32)
#define NCHUNK   (NSTEPS / CHUNK)
#define GRID_RES 128
#define RAYS_PER_BLOCK 8       // 8 waves * 32 lanes = 256 threads

// per-wave, per-buffer LDS layout (floats):
//   [0,32)    sigma
//   [32,64)   sun_v
//   [64,160)  rgb   (96)
//   [160,256) sky   (96)
#define BUF_FLOATS 256

// ---- gfx1250 async global->LDS 16B/lane copy --------------------------------
typedef int v4i __attribute__((vector_size(16)));
typedef __attribute__((address_space(1))) v4i* g_v4i_ptr;
typedef __attribute__((address_space(3))) v4i* l_v4i_ptr;

__device__ __forceinline__ void async_copy16(const float* g, float* l) {
#if __has_builtin(__builtin_amdgcn_global_load_async_to_lds_b128)
  __builtin_amdgcn_global_load_async_to_lds_b128(
      (g_v4i_ptr)g, (l_v4i_ptr)l, /*offset=*/0, /*cpol=*/0);
#else
  unsigned loff = (unsigned)(unsigned long long)
      (__attribute__((address_space(3))) void*)l;
  asm volatile("global_load_async_to_lds_b128 %0, %1, off"
               :: "v"(loff), "v"(g) : "memory");
#endif
}

#define WAIT_ASYNC_LE4() asm volatile("s_wait_asynccnt 0x4" ::: "memory")
#define WAIT_ASYNC_0()   asm volatile("s_wait_asynccnt 0x0" ::: "memory")
#define WAIT_DS_0()      asm volatile("s_wait_dscnt 0x0"    ::: "memory")

__global__ __launch_bounds__(256)
void nerf_render_kernel(const float* __restrict__ rays_o,
                        const float* __restrict__ rays_d,
                        const float* __restrict__ sigmas,
                        const float* __restrict__ rgbs,
                        const float* __restrict__ sun_v,
                        const float* __restrict__ sky,
                        const float* __restrict__ nears,
                        const float* __restrict__ fars,
                        const float* __restrict__ dgrid,
                        float* __restrict__ out,
                        int N) {
  __shared__ __align__(16) float smem[RAYS_PER_BLOCK * 2 * BUF_FLOATS]; // 16 KB

  const int lane = threadIdx.x & 31;
  const int wave = threadIdx.x >> 5;
  const int n    = blockIdx.x * RAYS_PER_BLOCK + wave;
  if (n >= N) return;

  float* buf0 = smem + wave * (2 * BUF_FLOATS);
  float* buf1 = buf0 + BUF_FLOATS;

  const float tnear = nears[n];
  const float tfar  = fars[n];
  const float dt    = (tfar - tnear) * (1.0f / (float)NSTEPS);
  const float ox = rays_o[3 * n + 0], oy = rays_o[3 * n + 1], oz = rays_o[3 * n + 2];
  const float dx = rays_d[3 * n + 0], dy = rays_d[3 * n + 1], dz = rays_d[3 * n + 2];

  const long long rayBase = (long long)n * NSTEPS;

  // ---- async stage one 32-step chunk into an LDS buffer (4 b128 async ops) --
  auto issue = [&](int base, float* buf) {
    long long rb = rayBase + base;
    if (lane < 8) {
      async_copy16(sigmas + rb + lane * 4, buf + lane * 4);            // 128 B
      async_copy16(sun_v  + rb + lane * 4, buf + 32 + lane * 4);       // 128 B
    }
    if (lane < 24) {
      async_copy16(rgbs + rb * 3 + lane * 4, buf + 64  + lane * 4);    // 384 B
      async_copy16(sky  + rb * 3 + lane * 4, buf + 160 + lane * 4);    // 384 B
    }
  };

  float carry = 0.0f;                 // running optical depth before this chunk
  float accW = 0.0f, accD = 0.0f, accR = 0.0f, accG = 0.0f, accB = 0.0f;

  // ---- consume one staged chunk: shade + scan + composite -------------------
  auto consume = [&](int base, const float* buf) {
    const float sig = buf[lane];
    const float sv  = buf[32 + lane];
    const float r0 = buf[64 + 3 * lane + 0];
    const float r1 = buf[64 + 3 * lane + 1];
    const float r2 = buf[64 + 3 * lane + 2];
    const float k0 = buf[160 + 3 * lane + 0];
    const float k1 = buf[160 + 3 * lane + 1];
    const float k2 = buf[160 + 3 * lane + 2];

    const float t = tnear + ((float)(base + lane) + 0.5f) * dt;
    const float x = ox + t * dx, y = oy + t * dy, z = oz + t * dz;
    const bool inside = (fabsf(x) < 1.0f) & (fabsf(y) < 1.0f) & (fabsf(z) < 1.0f);

    float sg = 0.0f;
    if (inside) {                      // grid value unused when outside AABB
      int ix = min(max((int)((x + 1.0f) * 64.0f), 0), GRID_RES - 1);
      int iy = min(max((int)((y + 1.0f) * 64.0f), 0), GRID_RES - 1);
      int iz = min(max((int)((z + 1.0f) * 64.0f), 0), GRID_RES - 1);
      float dg = dgrid[(ix << 14) | (iy << 7) | iz];
      if (dg > 0.01f) sg = sig;        // DENSITY_SCALE == 1
    }
    const float tau = sg * dt;

    // inclusive prefix sum of tau across the wave
    float incl = tau;
#pragma unroll
    for (int off = 1; off < 32; off <<= 1) {
      float p = __shfl_up(incl, off, 32);
      incl += (lane >= off) ? p : 0.0f;
    }

    const float T     = __expf(-(carry + incl - tau));   // exclusive transmittance
    const float alpha = 1.0f - __expf(-tau);
    const float w     = (T > 1e-3f) ? alpha * T : 0.0f;  // early-termination mask

    accW += w;
    accD += w * t;
    accR += w * (r0 * (sv + (1.0f - sv) * k0));
    accG += w * (r1 * (sv + (1.0f - sv) * k1));
    accB += w * (r2 * (sv + (1.0f - sv) * k2));

    carry += __shfl(incl, 31, 32);                       // chunk total
  };

  // ---- double-buffered async pipeline over 4 chunks -------------------------
  issue(0 * CHUNK, buf0);
  issue(1 * CHUNK, buf1);            // ASYNCcnt = 8

  WAIT_ASYNC_LE4();                  // chunk 0 landed (async loads in-order)
  consume(0 * CHUNK, buf0);
  WAIT_DS_0();                       // LDS reads of buf0 retired
  issue(2 * CHUNK, buf0);

  WAIT_ASYNC_LE4();                  // chunk 1 landed
  consume(1 * CHUNK, buf1);
  WAIT_DS_0();
  issue(3 * CHUNK, buf1);

  WAIT_ASYNC_LE4();                  // chunk 2 landed
  consume(2 * CHUNK, buf0);

  WAIT_ASYNC_0();                    // chunk 3 landed
  consume(3 * CHUNK, buf1);

  // ---- wave reduction (5 accumulators) --------------------------------------
#pragma unroll
  for (int off = 16; off > 0; off >>= 1) {
    accW += __shfl_down(accW, off, 32);
    accD += __shfl_down(accD, off, 32);
    accR += __shfl_down(accR, off, 32);
    accG += __shfl_down(accG, off, 32);
    accB += __shfl_down(accB, off, 32);
  }

  if (lane == 0) {
    const float bg = 1.0f - accW;                  // BG_COLOR = 1
    out[3 * n + 0] = accR + bg;
    out[3 * n + 1] = accG + bg;
    out[3 * n + 2] = accB + bg;
    out[3 * N + n] = fmaxf(accD - tnear, 0.0f) / (tfar - tnear);
    out[4 * N + n] = accW;
  }
}

extern "C" void kernel_launch(void* const* d_in, const int* in_sizes, int n_in,
                              void* d_out, int out_size, void* d_ws, size_t ws_size,
                              hipStream_t stream) {
  const float* rays_o = (const float*)d_in[0];
  const float* rays_d = (const float*)d_in[1];
  const float* sigmas = (const float*)d_in[2];
  const float* rgbs   = (const float*)d_in[3];
  const float* sun_v  = (const float*)d_in[4];
  const float* sky    = (const float*)d_in[5];
  const float* nears  = (const float*)d_in[6];
  const float* fars   = (const float*)d_in[7];
  const float* dgrid  = (const float*)d_in[8];
  float* out = (float*)d_out;

  const int N = in_sizes[6];                       // #rays (nears)
  dim3 block(256);
  dim3 grid((N + RAYS_PER_BLOCK - 1) / RAYS_PER_BLOCK);
  nerf_render_kernel<<<grid, block, 0, stream>>>(rays_o, rays_d, sigmas, rgbs,
                                                 sun_v, sky, nears, fars,
                                                 dgrid, out, N);
}